// PolarSCDecoder_61727269978287
// MI455X (gfx1250) — compile-verified
//
#include <hip/hip_runtime.h>
#include <stdint.h>

// Polar SC decoder, N=1024, frozen = first 512 positions, info = last 512.
// Closed-form reduction (see derivation): out[b,:] = F^{(x)9} (GF(2) butterfly)
// applied to c[i] = (llr_ch[b,i] + llr_ch[b,i+512] >= 0) ? 1 : 0.
// Pure streaming kernel: 64MB in, 32MB out -> ~4.1us at 23.3 TB/s.
// CDNA5 path: async global->LDS staging (ASYNCcnt) with per-wave double buffer,
// wave32 shuffle butterfly for the cross-lane XOR stages.

#define WAVES_PER_BLOCK 4
#define ROWS_PER_WAVE 4
#define NN 1024
#define HALF 512

__global__ __launch_bounds__(WAVES_PER_BLOCK * 32)
void polar_sc_closed_form(const float* __restrict__ llr,
                          float* __restrict__ out,
                          int batch) {
  __shared__ float buf[WAVES_PER_BLOCK][2][NN];  // 2 x 4KB per wave

  const int lane = threadIdx.x & 31;
  const int wv   = threadIdx.x >> 5;
  const int wave_global = blockIdx.x * WAVES_PER_BLOCK + wv;
  const int row0 = wave_global * ROWS_PER_WAVE;

  // LDS byte offsets of the two buffers (low 32 bits of the generic address
  // are the LDS byte address on gfx1250).
  const uint32_t lds_base[2] = {
      (uint32_t)(uintptr_t)&buf[wv][0][0],
      (uint32_t)(uintptr_t)&buf[wv][1][0]};

  // Stage one full 4KB row into LDS with 8 async b128 copies (512B/instr,
  // 16B per lane). GVS addressing: saddr = llr base, vaddr = 32-bit offset.
  auto issue_async_row = [&](int row, uint32_t ldsb) {
    uint32_t voff  = (uint32_t)row * (NN * 4u) + (uint32_t)lane * 16u;
    uint32_t laddr = ldsb + (uint32_t)lane * 16u;
#pragma unroll
    for (int j = 0; j < 8; ++j) {
      uint32_t vo = voff + 512u * (uint32_t)j;
      uint32_t la = laddr + 512u * (uint32_t)j;
      asm volatile("global_load_async_to_lds_b128 %0, %1, %2 offset:0"
                   :: "v"(la), "v"(vo), "s"(llr)
                   : "memory");
    }
  };

  if (row0 < batch) issue_async_row(row0, lds_base[0]);

#pragma unroll
  for (int k = 0; k < ROWS_PER_WAVE; ++k) {
    const int row = row0 + k;

    // Prefetch next row into the other buffer, then wait for current row.
    if (k + 1 < ROWS_PER_WAVE) {
      if (row0 + k + 1 < batch) issue_async_row(row0 + k + 1, lds_base[(k + 1) & 1]);
      asm volatile("s_wait_asynccnt 8" ::: "memory");  // first 8 (current row) done
    } else {
      asm volatile("s_wait_asynccnt 0" ::: "memory");
    }

    if (row >= batch) continue;

    // s[i] = llr[b,i] + llr[b,i+512]; bit c[i] = (s >= 0). Lane L owns
    // i in [16L, 16L+16) as bits 0..15 of m.
    const float4* pa =
        reinterpret_cast<const float4*>(&buf[wv][k & 1][lane * 16]);
    const float4* pb =
        reinterpret_cast<const float4*>(&buf[wv][k & 1][HALF + lane * 16]);
    unsigned m = 0u;
#pragma unroll
    for (int q = 0; q < 4; ++q) {
      float4 a = pa[q];
      float4 b = pb[q];
      m |= ((a.x + b.x >= 0.0f) ? 1u : 0u) << (4 * q + 0);
      m |= ((a.y + b.y >= 0.0f) ? 1u : 0u) << (4 * q + 1);
      m |= ((a.z + b.z >= 0.0f) ? 1u : 0u) << (4 * q + 2);
      m |= ((a.w + b.w >= 0.0f) ? 1u : 0u) << (4 * q + 3);
    }

    // 9-stage GF(2) butterfly over 512 bits: element i (i & span)==0 takes
    // XOR from i+span. Spans 256..16 -> cross-lane (distance 16..1),
    // spans 8..1 -> intra-register.
#pragma unroll
    for (int d = 16; d >= 1; d >>= 1) {
      unsigned v = __shfl_xor(m, d, 32);
      m ^= ((lane & d) == 0) ? v : 0u;
    }
    m ^= (m >> 8) & 0x00FFu;
    m ^= (m >> 4) & 0x0F0Fu;
    m ^= (m >> 2) & 0x3333u;
    m ^= (m >> 1) & 0x5555u;

    // Emit 16 floats (0.0/1.0) per lane, coalesced float4 stores.
    float4* po = reinterpret_cast<float4*>(out + (size_t)row * HALF + lane * 16);
#pragma unroll
    for (int q = 0; q < 4; ++q) {
      float4 o;
      o.x = ((m >> (4 * q + 0)) & 1u) ? 1.0f : 0.0f;
      o.y = ((m >> (4 * q + 1)) & 1u) ? 1.0f : 0.0f;
      o.z = ((m >> (4 * q + 2)) & 1u) ? 1.0f : 0.0f;
      o.w = ((m >> (4 * q + 3)) & 1u) ? 1.0f : 0.0f;
      po[q] = o;
    }
  }
}

extern "C" void kernel_launch(void* const* d_in, const int* in_sizes, int n_in,
                              void* d_out, int out_size, void* d_ws, size_t ws_size,
                              hipStream_t stream) {
  (void)n_in; (void)d_ws; (void)ws_size; (void)out_size;
  const float* llr = (const float*)d_in[0];
  float* out = (float*)d_out;
  const int batch = in_sizes[0] / NN;  // 16384
  const int rows_per_block = WAVES_PER_BLOCK * ROWS_PER_WAVE;  // 16
  const int grid = (batch + rows_per_block - 1) / rows_per_block;  // 1024
  polar_sc_closed_form<<<grid, WAVES_PER_BLOCK * 32, 0, stream>>>(llr, out, batch);
}